// Emb_loc_84696755077773
// MI455X (gfx1250) — compile-verified
//
#include <hip/hip_runtime.h>
#include <hip/hip_bf16.h>

// Problem constants (match reference)
#define NN 262144
#define PP 2500
#define KK 10
#define VV 11
#define EE 128
#define ALPHA 0.9f

typedef __attribute__((ext_vector_type(2))) float v2f;
typedef __attribute__((ext_vector_type(8))) float v8f;

// Precomputed per-location output table: table[p][e] = 0.9*loc + 0.1*sum(p^2)/sum(p)
__device__ float g_table[PP * EE];

// ---------------------------------------------------------------------------
// Kernel 1: precompute g_table via WMMA.
//   grid = ceil(P/16) blocks, block = 256 threads (8 waves).
//   Each block handles a 16-row (location) tile; wave w handles E-columns
//   [16w, 16w+16). Two GEMM accumulations (expw and expw^2) with
//   V_WMMA_F32_16X16X4_F32, K padded 11 -> 16 (4 k-chunks of 4).
// ---------------------------------------------------------------------------
__global__ void precompute_table_kernel(const int* __restrict__ poi_table,
                                        const float* __restrict__ emb_poi,
                                        const float* __restrict__ emb_loc) {
    __shared__ float s_expw[16 * 132];   // exp(emb_poi), rows padded to 16, pitch 132
    __shared__ float s_expw2[16 * 132];  // squared
    __shared__ float s_cnt[16 * 17];     // counts A-tile, pitch 17 (bank-friendly)

    const int tid = threadIdx.x;
    const int m0 = blockIdx.x * 16;

    // Stage exp tables into LDS (rows >= V are zero so padded K contributes 0).
    for (int i = tid; i < 16 * EE; i += 256) {
        int v = i >> 7;          // row
        int e = i & (EE - 1);    // col
        float w = (v < VV) ? __expf(emb_poi[v * EE + e]) : 0.0f;
        s_expw[v * 132 + e]  = w;
        s_expw2[v * 132 + e] = w * w;
    }

    // Category counts: thread t builds the count row for location m0+t.
    if (tid < 16) {
        int p = m0 + tid;
        int cat[KK];
#pragma unroll
        for (int k = 0; k < KK; ++k)
            cat[k] = (p < PP) ? poi_table[p * KK + k] : -1;
#pragma unroll
        for (int v = 0; v < 16; ++v) {
            int c = 0;
#pragma unroll
            for (int k = 0; k < KK; ++k) c += (cat[k] == v) ? 1 : 0;
            s_cnt[tid * 17 + v] = (float)c;
        }
    }
    __syncthreads();

    const int wave = tid >> 5;        // E-tile: columns [16*wave, 16*wave+16)
    const int lane = tid & 31;
    const int row  = lane & 15;
    const int hi   = lane >> 4;       // 0: lanes 0-15, 1: lanes 16-31
    const int nbase = wave * 16;
    const int col   = nbase + row;    // column this lane touches in B/C/D

    v8f ds = {};   // accumulates S  = C @ expw
    v8f dq = {};   // accumulates Q  = C @ expw^2

#pragma unroll
    for (int c = 0; c < 4; ++c) {
        // A 16x4 fp32 layout: lane<16 -> {K=4c+0, 4c+1}; lane>=16 -> {K=4c+2, 4c+3}
        const int k0 = 4 * c + 2 * hi;
        v2f a, bs, bq;
        a[0] = s_cnt[row * 17 + k0];
        a[1] = s_cnt[row * 17 + k0 + 1];
        // B 4x16 fp32 layout mirrors C/D striping: VGPR j holds rows {j, j+2}
        bs[0] = s_expw[(k0 + 0) * 132 + col];
        bs[1] = s_expw[(k0 + 1) * 132 + col];
        bq[0] = s_expw2[(k0 + 0) * 132 + col];
        bq[1] = s_expw2[(k0 + 1) * 132 + col];

        ds = __builtin_amdgcn_wmma_f32_16x16x4_f32(false, a, false, bs,
                                                   (short)0, ds, false, false);
        dq = __builtin_amdgcn_wmma_f32_16x16x4_f32(false, a, false, bq,
                                                   (short)0, dq, false, false);
    }

    // C/D layout: VGPR j, lanes 0-15 -> M=j, lanes 16-31 -> M=j+8; N=lane%16.
#pragma unroll
    for (int j = 0; j < 8; ++j) {
        int m = m0 + j + 8 * hi;
        if (m < PP) {
            float sv = ds[j];
            float qv = dq[j];
            float pv = qv / sv;
            g_table[m * EE + col] =
                emb_loc[m * EE + col] * ALPHA + pv * (1.0f - ALPHA);
        }
    }
}

// ---------------------------------------------------------------------------
// Kernel 2: pure row gather, the bandwidth-bound part (~128 MiB stores).
//   wave32: one wave moves exactly one 128-float row as 32 x b128.
//   Table (1.25 MB) lives entirely in L2, so reads are L2 hits.
// ---------------------------------------------------------------------------
__global__ void gather_rows_kernel(const int* __restrict__ x,
                                   float* __restrict__ out,
                                   int n_total) {
    const int wave = threadIdx.x >> 5;
    const int lane = threadIdx.x & 31;
    const long long n = (long long)blockIdx.x * 8 + wave;
    if (n >= n_total) return;

    const int p = x[n];
    const float4* __restrict__ src =
        reinterpret_cast<const float4*>(g_table + (long long)p * EE);
    float4* __restrict__ dst = reinterpret_cast<float4*>(out + n * EE);
    dst[lane] = src[lane];
}

extern "C" void kernel_launch(void* const* d_in, const int* in_sizes, int n_in,
                              void* d_out, int out_size, void* d_ws, size_t ws_size,
                              hipStream_t stream) {
    const int*   x        = (const int*)d_in[0];   // [N]
    const int*   poi      = (const int*)d_in[1];   // [P,K]
    const float* emb_poi  = (const float*)d_in[2]; // [V,E]
    const float* emb_loc  = (const float*)d_in[3]; // [P,E]
    float*       out      = (float*)d_out;         // [N,E]

    const int n_total = in_sizes[0];

    // Phase 1: build the 2500x128 fused table with WMMA (157 blocks).
    const int m_tiles = (PP + 15) / 16;
    precompute_table_kernel<<<m_tiles, 256, 0, stream>>>(poi, emb_poi, emb_loc);

    // Phase 2: bandwidth-bound gather; 8 rows per 256-thread block.
    const int blocks = (n_total + 7) / 8;
    gather_rows_kernel<<<blocks, 256, 0, stream>>>(x, out, n_total);
}